// AUTOGCNLayer_19576460935801
// MI455X (gfx1250) — compile-verified
//
#include <hip/hip_runtime.h>
#include <math.h>

typedef __attribute__((ext_vector_type(2))) float v2f;
typedef __attribute__((ext_vector_type(8))) float v8f;

#define DDIM   128
#define LDS_RS 132          // padded row stride (floats): 132 mod 64 = 4 -> conflict-free
#define KGAM   8
#define EPSC   1e-9f

// ---------------- utility kernels ----------------

__global__ void zero_kernel(float4* __restrict__ p, long n4) {
    long i = (long)blockIdx.x * blockDim.x + threadIdx.x;
    long stride = (long)gridDim.x * blockDim.x;
    float4 z; z.x = z.y = z.z = z.w = 0.0f;
    for (; i < n4; i += stride) p[i] = z;
}

// coef[0..5] = a0, b0, a1, b1, s2, c2
__global__ void coef_kernel(const float* __restrict__ gl_, const float* __restrict__ gh_,
                            const float* __restrict__ gm_, float* __restrict__ coef) {
    if (threadIdx.x == 0 && blockIdx.x == 0) {
        float a0 = 0.f, b0 = 0.f, a1 = 0.f, b1 = 0.f, s2 = 0.f, c2 = 0.f;
        for (int j = 0; j < KGAM; ++j) {
            float alpha = EPSC + (float)j * (1.0f - 2.0f * EPSC) / (float)(KGAM - 1);
            float malp  = EPSC + (float)j * (1.0f - EPSC) / (float)(KGAM - 1);
            float gl = fmaxf(gl_[j], 0.f);
            float gh = fmaxf(gh_[j], 0.f);
            float gm = fmaxf(gm_[j], 0.f);
            a0 += alpha * gl;         b0 += (1.0f - alpha) * gl;
            a1 += -alpha * gh;        b1 += (1.0f - alpha) * gh;
            s2 += gm;                 c2 += -malp * gm;
        }
        coef[0] = a0; coef[1] = b0; coef[2] = a1;
        coef[3] = b1; coef[4] = s2; coef[5] = c2;
    }
}

__global__ void deg_kernel(const int* __restrict__ dst, float* __restrict__ deg, int E) {
    int e = blockIdx.x * blockDim.x + threadIdx.x;
    if (e < E) atomicAdd(&deg[dst[e]], 1.0f);
}

__global__ void norm_kernel(float* __restrict__ deg_norm, int N) {
    int i = blockIdx.x * blockDim.x + threadIdx.x;
    if (i < N) deg_norm[i] = 1.0f / sqrtf(fmaxf(deg_norm[i], 1.0f));
}

// one wave per edge; lane handles 4 consecutive floats (32*4 = 128 = D)
// out[dst] += feat[src] * (norm[src]^npow); npow=1 for pass1 (feat raw),
// npow=2 for pass2 (feat = raw agg1, folds h = norm*agg1 and the src-side norm).
template <int NPOW>
__global__ __launch_bounds__(256) void spmm_kernel(const float* __restrict__ feat,
                                                   const float* __restrict__ norm,
                                                   const int* __restrict__ src,
                                                   const int* __restrict__ dst,
                                                   float* out, int E) {
    long gt = (long)blockIdx.x * blockDim.x + threadIdx.x;
    int e = (int)(gt >> 5);
    if (e >= E) return;
    int lane = (int)(gt & 31);
    int s = src[e], d = dst[e];
    float ns = norm[s];
    if (NPOW == 2) ns *= ns;
    const float4 v = *(const float4*)(feat + (long)s * DDIM + lane * 4);
    float* o = out + (long)d * DDIM + lane * 4;
    atomicAdd(o + 0, v.x * ns);
    atomicAdd(o + 1, v.y * ns);
    atomicAdd(o + 2, v.z * ns);
    atomicAdd(o + 3, v.w * ns);
}

// ---------------- fused triple GEMM (WMMA f32 16x16x4) + gating epilogue ----
// h  = norm[row]*agg1[row], h1 = norm[row]*agg2[row]  (folded into A coeffs)
// o0 = (ca0*h + cb0*f) @ Wl^T
// o1 = (ca1*h + cb1*f) @ Wh^T
// o2 = (cs2*h1 + cc2*f) @ Wm^T
// then sequential mutual sigmoid gating, +bias, *snorm, relu -> out.
// One wave per 16-row node tile; all three 128x128 W matrices live in LDS
// (3*128*132*4 = 198KB of the 320KB WGP LDS), rows padded to kill bank conflicts.
__device__ __forceinline__ float sigmoidf_(float x) { return 1.0f / (1.0f + expf(-x)); }

__global__ __launch_bounds__(256) void gemm_gate_kernel(const float* __restrict__ feat,
                                                        const float* __restrict__ agg1,
                                                        const float* __restrict__ agg2,
                                                        const float* __restrict__ Wl,
                                                        const float* __restrict__ Wh,
                                                        const float* __restrict__ Wm,
                                                        const float* __restrict__ coef,
                                                        const float* __restrict__ norm,
                                                        const float* __restrict__ bias,
                                                        const float* __restrict__ snorm,
                                                        float* __restrict__ out,
                                                        int n_tiles) {
    extern __shared__ float lds[];
    const int tid = threadIdx.x;

    // preload weights row-major with padded stride
    {
        const float* Ws[3] = {Wl, Wh, Wm};
        for (int g = 0; g < 3; ++g) {
            float* dl = lds + (long)g * DDIM * LDS_RS;
            const float* sw = Ws[g];
            for (int i = tid; i < DDIM * DDIM; i += 256) {
                int j = i >> 7, k = i & 127;
                dl[j * LDS_RS + k] = sw[i];
            }
        }
    }
    __syncthreads();

    const int wave = tid >> 5;
    const int lane = tid & 31;
    const int tile = blockIdx.x * 8 + wave;
    if (tile >= n_tiles) return;          // wave-uniform; EXEC stays all-ones for WMMA

    const int m  = lane & 15;             // A row / B col / D col within 16-tile
    const int kh = lane >> 4;             // selects k-pair {2kh, 2kh+1} inside each x4 step
    const long row = (long)tile * 16 + m;

    const float* h_row  = agg1 + row * DDIM + 2 * kh;
    const float* f_row  = feat + row * DDIM + 2 * kh;
    const float* h1_row = agg2 + row * DDIM + 2 * kh;

    const float nr  = norm[row];          // per-lane (per A-row) norm, folded into coeffs
    const float cA0 = coef[0] * nr, cB0 = coef[1];
    const float cA1 = coef[2] * nr, cB1 = coef[3];
    const float cS2 = coef[4] * nr, cC2 = coef[5];

    const v8f vz = {0.f, 0.f, 0.f, 0.f, 0.f, 0.f, 0.f, 0.f};
    const float* ldsB0 = lds;
    const float* ldsB1 = lds + (long)DDIM * LDS_RS;
    const float* ldsB2 = lds + 2L * DDIM * LDS_RS;

    v8f acc0[8], acc1[8], acc2[8];
#pragma unroll
    for (int nt = 0; nt < 8; ++nt) { acc0[nt] = vz; acc1[nt] = vz; acc2[nt] = vz; }

    // ---- GEMM0 + GEMM1 share agg1/f loads ----
    for (int ks = 0; ks < 32; ++ks) {
        v2f hv = *(const v2f*)(h_row + ks * 4);
        v2f fv = *(const v2f*)(f_row + ks * 4);
        v2f a0; a0.x = cA0 * hv.x + cB0 * fv.x; a0.y = cA0 * hv.y + cB0 * fv.y;
        v2f a1; a1.x = cA1 * hv.x + cB1 * fv.x; a1.y = cA1 * hv.y + cB1 * fv.y;
        const int bofs = ks * 4 + 2 * kh;
#pragma unroll
        for (int nt = 0; nt < 8; ++nt) {
            const int nrow = nt * 16 + m;
            v2f b0 = *(const v2f*)(ldsB0 + nrow * LDS_RS + bofs);
            v2f b1 = *(const v2f*)(ldsB1 + nrow * LDS_RS + bofs);
            acc0[nt] = __builtin_amdgcn_wmma_f32_16x16x4_f32(false, a0, false, b0,
                                                             (short)0, acc0[nt], false, false);
            acc1[nt] = __builtin_amdgcn_wmma_f32_16x16x4_f32(false, a1, false, b1,
                                                             (short)0, acc1[nt], false, false);
        }
    }

    // ---- GEMM2: agg2/f ----
    for (int ks = 0; ks < 32; ++ks) {
        v2f hv = *(const v2f*)(h1_row + ks * 4);
        v2f fv = *(const v2f*)(f_row + ks * 4);
        v2f a2; a2.x = cS2 * hv.x + cC2 * fv.x; a2.y = cS2 * hv.y + cC2 * fv.y;
        const int bofs = ks * 4 + 2 * kh;
#pragma unroll
        for (int nt = 0; nt < 8; ++nt) {
            const int nrow = nt * 16 + m;
            v2f b2 = *(const v2f*)(ldsB2 + nrow * LDS_RS + bofs);
            acc2[nt] = __builtin_amdgcn_wmma_f32_16x16x4_f32(false, a2, false, b2,
                                                             (short)0, acc2[nt], false, false);
        }
    }

    // ---- in-register gating epilogue ----
    // D layout: lane L -> col L%16, VGPR r -> row r + 8*(L/16)
    float sn[8];
#pragma unroll
    for (int r = 0; r < 8; ++r) sn[r] = snorm[(long)tile * 16 + r + 8 * kh];

#pragma unroll
    for (int nt = 0; nt < 8; ++nt) {
        const int col = nt * 16 + m;
        const float bia = bias[col];
#pragma unroll
        for (int r = 0; r < 8; ++r) {
            float o0 = acc0[nt][r], o1 = acc1[nt][r], o2 = acc2[nt][r];
            o0 = o0 * sigmoidf_(o1 + o2);
            o1 = o1 * sigmoidf_(o0 + o2);
            o2 = o2 * sigmoidf_(o0 + o1);
            const long orow = (long)tile * 16 + r + 8 * kh;
            const float v = (o0 + o1 + o2 + bia) * sn[r];
            out[orow * DDIM + col] = fmaxf(v, 0.0f);
        }
    }
}

// ---------------- host launcher ----------------
extern "C" void kernel_launch(void* const* d_in, const int* in_sizes, int n_in,
                              void* d_out, int out_size, void* d_ws, size_t ws_size,
                              hipStream_t stream) {
    (void)n_in; (void)out_size; (void)ws_size;
    const float* feature = (const float*)d_in[0];
    const float* snorm   = (const float*)d_in[1];
    const int*   src     = (const int*)d_in[2];
    const int*   dst     = (const int*)d_in[3];
    const float* W_low   = (const float*)d_in[4];
    const float* W_high  = (const float*)d_in[5];
    const float* W_mid   = (const float*)d_in[6];
    const float* g_low   = (const float*)d_in[7];
    const float* g_high  = (const float*)d_in[8];
    const float* g_mid   = (const float*)d_in[9];
    const float* bias    = (const float*)d_in[10];
    float* out = (float*)d_out;

    const int  N  = in_sizes[1];          // snorm_n has N elements
    const int  E  = in_sizes[2];
    const long ND = (long)N * DDIM;

    float* wsf   = (float*)d_ws;
    float* coef  = wsf;                   // 16 floats
    float* norm  = wsf + 16;              // N floats (deg -> norm in place)
    float* bufA1 = wsf + 16 + N;          // N*D raw aggregate 1 (pre-norm h)
    float* bufA2 = bufA1 + ND;            // N*D raw aggregate 2 (pre-norm h1)

    // zero deg + both accumulators
    {
        long n4 = (16 + (long)N + 2 * ND + 3) / 4;
        long b  = (n4 + 255) / 256;
        int blocks = (int)(b < 4096 ? b : 4096);
        zero_kernel<<<blocks, 256, 0, stream>>>((float4*)wsf, n4);
    }
    coef_kernel<<<1, 32, 0, stream>>>(g_low, g_high, g_mid, coef);
    deg_kernel<<<(E + 255) / 256, 256, 0, stream>>>(dst, norm, E);
    norm_kernel<<<(N + 255) / 256, 256, 0, stream>>>(norm, N);

    {   // SpMM1: bufA1[dst] += feature[src] * norm[src]
        int blocks = (int)(((long)E * 32 + 255) / 256);
        spmm_kernel<1><<<blocks, 256, 0, stream>>>(feature, norm, src, dst, bufA1, E);
    }
    {   // SpMM2: bufA2[dst] += agg1[src] * norm[src]^2   (== h[src]*norm[src])
        int blocks = (int)(((long)E * 32 + 255) / 256);
        spmm_kernel<2><<<blocks, 256, 0, stream>>>(bufA1, norm, src, dst, bufA2, E);
    }
    {   // fused triple GEMM + gating + bias + graph-norm + relu -> d_out
        int n_tiles = N / 16;
        int blocks  = (n_tiles + 7) / 8;
        size_t lds_bytes = 3ull * DDIM * LDS_RS * sizeof(float);   // 202752 B of 320KB WGP LDS
        gemm_gate_kernel<<<blocks, 256, lds_bytes, stream>>>(feature, bufA1, bufA2,
                                                             W_low, W_high, W_mid, coef,
                                                             norm, bias, snorm, out, n_tiles);
    }
}